// KIARGCN_1056561954823
// MI455X (gfx1250) — compile-verified
//
#include <hip/hip_runtime.h>

#define N_NODES 100000
#define N_REL   4
#define N_EDGES 600000
#define D_FEAT  128

typedef __attribute__((ext_vector_type(2))) float v2f;
typedef __attribute__((ext_vector_type(8))) float v8f;

// ---------------------------------------------------------------------------
// Zero helpers (float4 stores; n must be a multiple of 4 for the f32x4 one)
// ---------------------------------------------------------------------------
__global__ void rgcn_zero_f32x4(float4* __restrict__ p, int n4) {
    int i = blockIdx.x * blockDim.x + threadIdx.x;
    if (i < n4) p[i] = make_float4(0.f, 0.f, 0.f, 0.f);
}
__global__ void rgcn_zero_i32(int* __restrict__ p, int n) {
    int i = blockIdx.x * blockDim.x + threadIdx.x;
    if (i < n) p[i] = 0;
}

// ---------------------------------------------------------------------------
// Degree histogram for all relations at once (int atomics, L2-resident)
// ---------------------------------------------------------------------------
__global__ void rgcn_degrees(const int* __restrict__ edges,
                             int* __restrict__ degOut, int* __restrict__ degIn) {
    int t = blockIdx.x * blockDim.x + threadIdx.x;
    if (t >= N_REL * N_EDGES) return;
    int r = t / N_EDGES;
    int e = t - r * N_EDGES;
    const int* base = edges + (size_t)r * 2 * N_EDGES;
    int s = base[e];
    int d = base[N_EDGES + e];
    atomicAdd(&degOut[r * N_NODES + s], 1);
    atomicAdd(&degIn [r * N_NODES + d], 1);
}

// ---------------------------------------------------------------------------
// SpMM scatter for one relation: Y[dst] += X[src] * rsqrt(deg_out[src])
// One wave (32 lanes) per edge; each lane moves a float4 (32*4 = 128 = D).
// X (51MB) and Y (51MB) both sit in the 192MB L2 -> gathers + f32 atomics
// run at L2 speed; this phase dominates runtime, not the GEMM.
// ---------------------------------------------------------------------------
__global__ void rgcn_scatter(const int* __restrict__ src, const int* __restrict__ dst,
                             const float* __restrict__ X, const int* __restrict__ degOut,
                             float* __restrict__ Y) {
    int t = blockIdx.x * blockDim.x + threadIdx.x;
    int e = t >> 5;
    int lane = t & 31;
    if (e >= N_EDGES) return;
    int s = src[e];
    int d = dst[e];
    int dg = degOut[s];
    float sc = rsqrtf((float)(dg < 1 ? 1 : dg));
    const float4 x = *(const float4*)(X + (size_t)s * D_FEAT + lane * 4);
    float* y = Y + (size_t)d * D_FEAT + lane * 4;
    atomicAdd(y + 0, x.x * sc);
    atomicAdd(y + 1, x.y * sc);
    atomicAdd(y + 2, x.z * sc);
    atomicAdd(y + 3, x.w * sc);
}

// ---------------------------------------------------------------------------
// Dense transform with f32 WMMA:
//   Z[m, :] += (Y[m, :] * rsqrt(deg_in[m])) @ W   (W is 128x128 for this rel)
//
// Block = 256 threads = 8 wave32s; each wave computes a 16-row x 128-col tile
// via 8 accumulators of V_WMMA_F32_16X16X4_F32. W is staged transposed in LDS
// in two 64-K chunks, padded stride 68 (68 % 64 == 4 => the float2 b-loads
// hit all 64 banks exactly once across the wave).
//
// ISA lane layouts (CDNA5 ISA):
//   A 16x4 f32 : lanes 0-15 -> M=lane, VGPR{0,1}=K{0,1}; lanes 16-31 -> K{2,3}
//   B 4x16 f32 : lanes 0-15 -> N=lane, VGPR{0,1}=K{0,1}; lanes 16-31 -> K{2,3}
//   C/D 16x16  : VGPR v, lanes 0-15 -> M=v, N=lane; lanes 16-31 -> M=v+8
//
// A loads are UNCONDITIONAL (pointer clamped to row 0, scale s=0 for invalid
// rows) so no exec-mask churn in the hot loop; the 8 b-operands are read into
// distinct registers before the WMMA burst so 8 ds_loads pipeline.
// ---------------------------------------------------------------------------
#define WT_STRIDE 68   // 64 K values + pad; 68 % 64 == 4 -> conflict-free b loads

__global__ __launch_bounds__(256)
void rgcn_gemm(const float* __restrict__ Y, const float* __restrict__ W,
               const int* __restrict__ degIn, float* __restrict__ Z) {
    __shared__ float wt[D_FEAT * WT_STRIDE];   // 128 * 68 * 4B = 34,816 B

    const int wave = threadIdx.x >> 5;         // 0..7
    const int lane = threadIdx.x & 31;
    const int half = lane >> 4;                // 0: K+{0,1}   1: K+{2,3}
    const int l16  = lane & 15;
    const int mbase = blockIdx.x * 128 + wave * 16;

    const int arow = mbase + l16;              // row this lane feeds into A
    const bool rowOK = arow < N_NODES;
    int dg = rowOK ? degIn[rowOK ? arow : 0] : 1;
    const float s = rowOK ? rsqrtf((float)(dg < 1 ? 1 : dg)) : 0.0f;
    const float* aptr = Y + (size_t)(rowOK ? arow : 0) * D_FEAT + half * 2;

    // C init from Z (accumulate across relations)
    v8f acc[8];
#pragma unroll
    for (int nt = 0; nt < 8; ++nt) {
#pragma unroll
        for (int v = 0; v < 8; ++v) {
            int m = mbase + v + half * 8;
            acc[nt][v] = (m < N_NODES) ? Z[(size_t)m * D_FEAT + nt * 16 + l16] : 0.0f;
        }
    }

    for (int chunk = 0; chunk < 2; ++chunk) {
        // Stage W[k=chunk*64 .. +63][n] transposed into LDS: wt[n*68 + klocal]
        __syncthreads();
        for (int i = threadIdx.x; i < 64 * D_FEAT; i += 256) {
            int kl = i >> 7;           // 0..63
            int n  = i & 127;
            wt[n * WT_STRIDE + kl] = W[(size_t)(chunk * 64 + kl) * D_FEAT + n];
        }
        __syncthreads();

        const float* ac = aptr + chunk * 64;
#pragma unroll 4
        for (int k = 0; k < 64; k += 4) {
            // Unconditional, clamped A load; invalid rows contribute 0 via s.
            v2f a = *(const v2f*)(ac + k);     // K = k+half*2, k+half*2+1
            a *= s;
            // Pull all 8 b tiles into flight before the WMMA burst.
            v2f b[8];
#pragma unroll
            for (int nt = 0; nt < 8; ++nt)
                b[nt] = *(const v2f*)(&wt[(nt * 16 + l16) * WT_STRIDE + k + half * 2]);
#pragma unroll
            for (int nt = 0; nt < 8; ++nt)
                acc[nt] = __builtin_amdgcn_wmma_f32_16x16x4_f32(
                    false, a, false, b[nt], (short)0, acc[nt], false, false);
        }
    }

    // Store D back to Z
#pragma unroll
    for (int nt = 0; nt < 8; ++nt) {
#pragma unroll
        for (int v = 0; v < 8; ++v) {
            int m = mbase + v + half * 8;
            if (m < N_NODES) Z[(size_t)m * D_FEAT + nt * 16 + l16] = acc[nt][v];
        }
    }
}

// ---------------------------------------------------------------------------
// Launch
// ---------------------------------------------------------------------------
extern "C" void kernel_launch(void* const* d_in, const int* in_sizes, int n_in,
                              void* d_out, int out_size, void* d_ws, size_t ws_size,
                              hipStream_t stream) {
    const int*   edges = (const int*)d_in[0];   // [4, 2, 600000]
    const float* X     = (const float*)d_in[1]; // [100000, 128]
    const float* W     = (const float*)d_in[2]; // [4, 128, 128]
    float* Z = (float*)d_out;                   // [100000, 128]

    // Workspace layout
    float* Yw     = (float*)d_ws;                               // 12.8M f32 (51.2 MB)
    int*   degOut = (int*)((char*)d_ws + (size_t)N_NODES * D_FEAT * 4);  // 400k
    int*   degIn  = degOut + N_REL * N_NODES;                   // 400k

    const int ZN  = N_NODES * D_FEAT;  // 12,800,000 (multiple of 4)
    const int ZN4 = ZN / 4;

    // 1) degrees
    rgcn_zero_i32<<<(2 * N_REL * N_NODES + 255) / 256, 256, 0, stream>>>(degOut,
                                                                         2 * N_REL * N_NODES);
    rgcn_degrees<<<(N_REL * N_EDGES + 255) / 256, 256, 0, stream>>>(edges, degOut, degIn);

    // 2) Z = 0
    rgcn_zero_f32x4<<<(ZN4 + 255) / 256, 256, 0, stream>>>((float4*)Z, ZN4);

    // 3) per relation: scatter then WMMA transform (accumulating into Z)
    const int scatterThreads = N_EDGES * 32;
    const int gemmBlocks = (N_NODES + 127) / 128;  // 782
    for (int r = 0; r < N_REL; ++r) {
        const int* src = edges + (size_t)r * 2 * N_EDGES;
        const int* dst = src + N_EDGES;
        rgcn_zero_f32x4<<<(ZN4 + 255) / 256, 256, 0, stream>>>((float4*)Yw, ZN4);
        rgcn_scatter<<<(scatterThreads + 255) / 256, 256, 0, stream>>>(
            src, dst, X, degOut + r * N_NODES, Yw);
        rgcn_gemm<<<gemmBlocks, 256, 0, stream>>>(
            Yw, W + (size_t)r * D_FEAT * D_FEAT, degIn + r * N_NODES, Z);
    }
}